// HybridSparseMaskGen3D_45183055954075
// MI455X (gfx1250) — compile-verified
//
#include <hip/hip_runtime.h>
#include <math.h>

// jnp.finfo(float32).min == -FLT_MAX (most-negative FINITE float, not -inf)
#define NEG_FILL (-3.4028234663852886e+38f)

typedef float v4f __attribute__((ext_vector_type(4)));
typedef int   v4i __attribute__((ext_vector_type(4)));

#if defined(__has_builtin)
#  if __has_builtin(__builtin_amdgcn_global_store_async_from_lds_b128)
#    define HAVE_ASYNC_STORE 1
#  endif
#  if __has_builtin(__builtin_amdgcn_s_wait_asynccnt)
#    define HAVE_WAIT_ASYNC 1
#  endif
#endif

// Builtin expects: (v4i __device__ *gaddr, v4i __shared__ *lds_src, imm, imm)
typedef __attribute__((address_space(1))) v4i as1_v4i;
typedef __attribute__((address_space(3))) v4i as3_v4i;

// ---------------------------------------------------------------------------
// Pass 1: fill entire N*N output with -FLT_MAX at store-bandwidth.
// CDNA5 path: each lane owns a 16B -FLT_MAX pattern slot in LDS and issues
// GLOBAL_STORE_ASYNC_FROM_LDS_B128 ops (ASYNCcnt-tracked DMA, no VGPR data
// movement). Fallback: non-temporal 128-bit stores (output > 192MB L2).
// ---------------------------------------------------------------------------
__global__ void HybridMask_fill_negmax(float* __restrict__ out,
                                       long long nvec, int ntail) {
  const int tid = threadIdx.x;
  __shared__ v4f pat[256];
  pat[tid] = (v4f){NEG_FILL, NEG_FILL, NEG_FILL, NEG_FILL};
  __syncthreads();  // DS writes visible (s_wait_dscnt) before async engine reads

  v4f* __restrict__ o4 = (v4f*)out;
  long long i = (long long)blockIdx.x * blockDim.x + tid;
  const long long stride = (long long)gridDim.x * blockDim.x;

#if defined(HAVE_ASYNC_STORE)
  for (; i < nvec; i += stride) {
    // 32 lanes x 16B = 512B contiguous per wave per issue; ASYNCcnt-tracked.
    __builtin_amdgcn_global_store_async_from_lds_b128(
        (as1_v4i*)(o4 + i), (as3_v4i*)&pat[tid], 0, 0);
  }
#  if defined(HAVE_WAIT_ASYNC)
  __builtin_amdgcn_s_wait_asynccnt(0);
#  endif
#else
  const v4f v = {NEG_FILL, NEG_FILL, NEG_FILL, NEG_FILL};
  for (; i < nvec; i += stride) {
    __builtin_nontemporal_store(v, o4 + i);
  }
#endif

  // Tail elements (none for N=8192, kept for generality).
  if (blockIdx.x == 0 && tid < ntail) out[nvec * 4 + tid] = NEG_FILL;
}

// ---------------------------------------------------------------------------
// Pass 2: scatter 0.0f at the <=143 unmasked columns of each row.
//   threads 0..124  : local window offsets (dt,dh,dw) in [-2,2]^3
//   threads 125..142: dilated offsets {+-1,+-2,+-4} along exactly one axis
// Duplicates (axis offsets |d|<=2 also inside window) are idempotent.
// ---------------------------------------------------------------------------
__global__ void HybridMask_scatter_zeros(const int* __restrict__ tp,
                                         const int* __restrict__ hp,
                                         const int* __restrict__ wp,
                                         float* __restrict__ out, int N) {
  const int t = tp[0], h = hp[0], w = wp[0];
  const int i = blockIdx.x;
  const int hw = h * w;
  const int ti = i / hw;
  const int r0 = i - ti * hw;
  const int hi = r0 / w;
  const int wi = r0 - hi * w;

  const int tid = threadIdx.x;
  int dt = 0, dh = 0, dw = 0;
  bool valid = false;
  if (tid < 125) {                       // 5*5*5 local window
    dt = tid / 25 - 2;
    const int r = tid % 25;
    dh = r / 5 - 2;
    dw = r % 5 - 2;
    valid = true;
  } else if (tid < 143) {                // 3 axes * {1,2,4,-1,-2,-4}
    const int k = tid - 125;
    const int axis = k / 6;
    const int s = k % 6;
    const int off = (s < 3) ? (1 << s) : -(1 << (s - 3));
    if (axis == 0)      dt = off;
    else if (axis == 1) dh = off;
    else                dw = off;
    valid = true;
  }
  if (valid) {
    const int tj = ti + dt, hj = hi + dh, wj = wi + dw;
    if ((unsigned)tj < (unsigned)t && (unsigned)hj < (unsigned)h &&
        (unsigned)wj < (unsigned)w) {
      const long long j = ((long long)tj * h + hj) * w + wj;
      out[(long long)i * N + j] = 0.0f;
    }
  }
}

extern "C" void kernel_launch(void* const* d_in, const int* in_sizes, int n_in,
                              void* d_out, int out_size, void* d_ws, size_t ws_size,
                              hipStream_t stream) {
  (void)in_sizes; (void)n_in; (void)d_ws; (void)ws_size;

  const long long total = (long long)out_size;          // N*N elements
  const int N = (int)(sqrt((double)total) + 0.5);       // host-side, no dev reads
  const long long nvec = total / 4;                     // # of 16B chunks
  const int ntail = (int)(total - nvec * 4);

  float* out = (float*)d_out;

  long long want = (nvec + 255) / 256;
  int blocks = (want > 32768) ? 32768 : (int)(want > 0 ? want : 1);
  HybridMask_fill_negmax<<<blocks, 256, 0, stream>>>(out, nvec, ntail);

  // Same stream => ordered after the fill.
  HybridMask_scatter_zeros<<<N, 160, 0, stream>>>(
      (const int*)d_in[0], (const int*)d_in[1], (const int*)d_in[2], out, N);
}